// YOLOv1Loss_26302379721165
// MI455X (gfx1250) — compile-verified
//
#include <hip/hip_runtime.h>

#define NGRID 7
#define NCLS 80
#define DCH 90                // 10 + 80
#define LAMBDA_COORD 5.0f
#define LAMBDA_NOOBJ 0.5f
#define EPS_IOU 1e-6f
#define EPS_SQRT 1e-6f
#define BLOCK 256

typedef float v2f __attribute__((ext_vector_type(2)));
typedef float v8f __attribute__((ext_vector_type(8)));

// ---------------------------------------------------------------------------
// IOU of two boxes already converted to image coords (cx, cy, w, h)
// ---------------------------------------------------------------------------
__device__ __forceinline__ float iou_img(float cx1, float cy1, float w1, float h1,
                                         float cx2, float cy2, float w2, float h2) {
    float ixmin = fmaxf(cx1 - w1 * 0.5f, cx2 - w2 * 0.5f);
    float iymin = fmaxf(cy1 - h1 * 0.5f, cy2 - h2 * 0.5f);
    float ixmax = fminf(cx1 + w1 * 0.5f, cx2 + w2 * 0.5f);
    float iymax = fminf(cy1 + h1 * 0.5f, cy2 + h2 * 0.5f);
    float iw = fmaxf(ixmax - ixmin, 0.0f);
    float ih = fmaxf(iymax - iymin, 0.0f);
    float inter = iw * ih;
    float uni   = w1 * h1 + w2 * h2 - inter;
    return inter / (uni + EPS_IOU);
}

// ---------------------------------------------------------------------------
// Stage 1: one thread per grid cell; per-block reduction to float4 partials
//   partial = {coord_sum, conf_sum, class_sum, n_obj}
// ---------------------------------------------------------------------------
__global__ void __launch_bounds__(BLOCK)
yolo_stage1(const float* __restrict__ pred,
            const float* __restrict__ target,
            float4* __restrict__ part, int ncells) {
    const int tid  = threadIdx.x;
    const int cell = blockIdx.x * BLOCK + tid;

    float coord = 0.0f, conf = 0.0f, cls = 0.0f, nobj = 0.0f;

    if (cell < ncells) {
        const float2* p2 = reinterpret_cast<const float2*>(pred   + (size_t)cell * DCH);
        const float2* t2 = reinterpret_cast<const float2*>(target + (size_t)cell * DCH);

        // pred[0..9]: x1 y1 w1 h1 c1 x2 y2 w2 h2 c2
        float2 pa = p2[0];   // x1 y1
        float2 pb = p2[1];   // w1 h1
        float2 pc = p2[2];   // c1 x2
        float2 pd = p2[3];   // y2 w2
        float2 pe = p2[4];   // h2 c2
        // target[0..5]: tx ty tw th tconf (t5 unused)
        float2 ta = t2[0];
        float2 tb = t2[1];
        float2 tc = t2[2];

        const float S  = (float)NGRID;
        const float gx = (float)(cell % NGRID);
        const float gy = (float)((cell / NGRID) % NGRID);

        float cx1 = (gx + pa.x) / S, cy1 = (gy + pa.y) / S;
        float cx2 = (gx + pc.y) / S, cy2 = (gy + pd.x) / S;
        float tcx = (gx + ta.x) / S, tcy = (gy + ta.y) / S;

        float iou1 = iou_img(cx1, cy1, pb.x, pb.y, tcx, tcy, tb.x, tb.y);
        float iou2 = iou_img(cx2, cy2, pd.y, pe.x, tcx, tcy, tb.x, tb.y);

        bool resp1 = iou1 > iou2;
        float bx = resp1 ? pa.x : pc.y;
        float by = resp1 ? pa.y : pd.x;
        float bw = resp1 ? pb.x : pd.y;
        float bh = resp1 ? pb.y : pe.x;
        float bc = resp1 ? pc.x : pe.y;

        bool obj = (tc.x == 1.0f);
        if (obj) {
            float dx = bx - ta.x, dy = by - ta.y;
            float dxy2 = dx * dx + dy * dy;
            float sw = sqrtf(fmaxf(bw, EPS_SQRT)) - sqrtf(fmaxf(tb.x, EPS_SQRT));
            float sh = sqrtf(fmaxf(bh, EPS_SQRT)) - sqrtf(fmaxf(tb.y, EPS_SQRT));
            float dwh2 = sw * sw + sh * sh;
            coord = 0.5f * (dxy2 + dwh2);

            float e = bc - 1.0f;
            conf = e * e;
            nobj = 1.0f;

            // class loss: mean over 80 of (p - t)^2  (only loaded when obj)
            float cs = 0.0f;
#pragma unroll 8
            for (int i = 5; i < 45; ++i) {
                float2 pp = p2[i];
                float2 tt = t2[i];
                float da = pp.x - tt.x;
                float db = pp.y - tt.y;
                cs += da * da + db * db;
            }
            cls = cs * (1.0f / (float)NCLS);
        } else {
            conf = LAMBDA_NOOBJ * (pc.x * pc.x + pe.y * pe.y);
        }
    }

    __shared__ float4 red[BLOCK];
    red[tid] = make_float4(coord, conf, cls, nobj);
    __syncthreads();
#pragma unroll
    for (int s = BLOCK / 2; s > 0; s >>= 1) {
        if (tid < s) {
            float4 a = red[tid], b = red[tid + s];
            red[tid] = make_float4(a.x + b.x, a.y + b.y, a.z + b.z, a.w + b.w);
        }
        __syncthreads();
    }
    if (tid == 0) part[blockIdx.x] = red[0];
}

// ---------------------------------------------------------------------------
// Full-wave (32 lane) sum via v_wmma_f32_16x16x4_f32 with a ones B-matrix.
// Level 1: A = (q, 0) -> D1[m,*] = q[m] + q[m+16]            (1 WMMA)
// Level 2: 4 accumulate-chained WMMAs over (d1[2p], d1[2p+1]) fold the 16
//          rows; every element of the result equals sum over all 32 lanes.
// Must be executed with EXEC all-1s within the wave.
// ---------------------------------------------------------------------------
__device__ __forceinline__ float wave32_sum_wmma(float q) {
    v2f ones; ones[0] = 1.0f; ones[1] = 1.0f;
    v2f a;    a[0] = q;       a[1] = 0.0f;
    v8f zero = {0.f, 0.f, 0.f, 0.f, 0.f, 0.f, 0.f, 0.f};

    v8f d1 = __builtin_amdgcn_wmma_f32_16x16x4_f32(
        false, a, false, ones, (short)0, zero, false, false);

    v8f acc = {0.f, 0.f, 0.f, 0.f, 0.f, 0.f, 0.f, 0.f};
    v2f a2;
    a2[0] = d1[0]; a2[1] = d1[1];
    acc = __builtin_amdgcn_wmma_f32_16x16x4_f32(false, a2, false, ones, (short)0, acc, false, false);
    a2[0] = d1[2]; a2[1] = d1[3];
    acc = __builtin_amdgcn_wmma_f32_16x16x4_f32(false, a2, false, ones, (short)0, acc, false, false);
    a2[0] = d1[4]; a2[1] = d1[5];
    acc = __builtin_amdgcn_wmma_f32_16x16x4_f32(false, a2, false, ones, (short)0, acc, false, false);
    a2[0] = d1[6]; a2[1] = d1[7];
    acc = __builtin_amdgcn_wmma_f32_16x16x4_f32(false, a2, false, ones, (short)0, acc, false, false);
    return acc[0];
}

// ---------------------------------------------------------------------------
// Stage 2: single block; deterministic reduction of per-block partials,
// final 32-lane fold done with WMMA, then scalar loss.
// ---------------------------------------------------------------------------
__global__ void __launch_bounds__(BLOCK)
yolo_stage2(const float4* __restrict__ part, int nblk,
            float ncellsf, float* __restrict__ out) {
    const int tid = threadIdx.x;

    float4 acc = make_float4(0.f, 0.f, 0.f, 0.f);
    for (int i = tid; i < nblk; i += BLOCK) {
        float4 v = part[i];
        acc.x += v.x; acc.y += v.y; acc.z += v.z; acc.w += v.w;
    }

    __shared__ float4 red[BLOCK];
    red[tid] = acc;
    __syncthreads();
#pragma unroll
    for (int s = BLOCK / 2; s >= 32; s >>= 1) {
        if (tid < s) {
            float4 a = red[tid], b = red[tid + s];
            red[tid] = make_float4(a.x + b.x, a.y + b.y, a.z + b.z, a.w + b.w);
        }
        __syncthreads();
    }

    if (tid < 32) {               // wave 0 only: EXEC all-1s inside this wave
        float4 q = red[tid];
        float coord_s = wave32_sum_wmma(q.x);
        float conf_s  = wave32_sum_wmma(q.y);
        float class_s = wave32_sum_wmma(q.z);
        float n_obj   = wave32_sum_wmma(q.w);

        if (tid == 0) {
            float denom      = fmaxf(n_obj, 1.0f);
            float conf_count = n_obj + 2.0f * (ncellsf - n_obj);
            float loss = LAMBDA_COORD * (coord_s / denom)
                       + conf_s / fmaxf(conf_count, 1.0f)
                       + class_s / denom;
            out[0] = loss;
        }
    }
}

// ---------------------------------------------------------------------------
extern "C" void kernel_launch(void* const* d_in, const int* in_sizes, int n_in,
                              void* d_out, int out_size, void* d_ws, size_t ws_size,
                              hipStream_t stream) {
    const float* pred   = (const float*)d_in[0];
    const float* target = (const float*)d_in[1];

    const int ncells = in_sizes[0] / DCH;          // B * 7 * 7 = 200704
    const int nblk   = (ncells + BLOCK - 1) / BLOCK;

    float4* part = (float4*)d_ws;                  // nblk * 16 bytes

    yolo_stage1<<<nblk, BLOCK, 0, stream>>>(pred, target, part, ncells);
    yolo_stage2<<<1, BLOCK, 0, stream>>>(part, nblk, (float)ncells, (float*)d_out);
}